// StronglyEntanglingConv2d_78658031058951
// MI455X (gfx1250) — compile-verified
//
#include <hip/hip_runtime.h>
#include <math.h>

// ---------------------------------------------------------------------------
// Stage 1: build U row-by-row (one workgroup per row, state in LDS)
// Stage 2: out = X * Ure^T (+ i * X * Uim^T) via V_WMMA_F32_16X16X4_F32
// ---------------------------------------------------------------------------

#define NQ 10
#define NL 5
#define NN 1024      // 2^NQ
#define MT 2048      // 128*16 batch rows

typedef __attribute__((ext_vector_type(2))) float v2f;
typedef __attribute__((ext_vector_type(8))) float v8f;

// ---------------- Kernel 1: circuit simulation of basis rows ----------------
__global__ __launch_bounds__(512)
void build_u_rows(const float* __restrict__ w,
                  float* __restrict__ Ure,
                  float* __restrict__ Uim) {
    __shared__ float sre[NN];
    __shared__ float sim[NN];
    __shared__ float g_re[NL * NQ][4];
    __shared__ float g_im[NL * NQ][4];

    const int t   = threadIdx.x;
    const int row = blockIdx.x;

    // Precompute the 50 U3 gates (threads 0..49)
    if (t < NL * NQ) {
        float th  = w[t * 3 + 0];
        float ph  = w[t * 3 + 1];
        float lam = w[t * 3 + 2];
        float c = cosf(0.5f * th);
        float s = sinf(0.5f * th);
        g_re[t][0] = c;                  g_im[t][0] = 0.0f;
        g_re[t][1] = -cosf(lam) * s;     g_im[t][1] = -sinf(lam) * s;
        g_re[t][2] =  cosf(ph) * s;      g_im[t][2] =  sinf(ph) * s;
        g_re[t][3] =  cosf(ph + lam)*c;  g_im[t][3] =  sinf(ph + lam)*c;
    }
    // state = e_row
    for (int j = t; j < NN; j += 512) {
        sre[j] = (j == row) ? 1.0f : 0.0f;
        sim[j] = 0.0f;
    }
    __syncthreads();

    for (int l = 0; l < NL; ++l) {
        // ---- single-qubit U3 gates (qubit q <-> bit position 9-q) ----
        for (int q = 0; q < NQ; ++q) {
            const int p  = (NQ - 1) - q;
            const int gi = l * NQ + q;
            const int j0 = ((t >> p) << (p + 1)) | (t & ((1 << p) - 1));
            const int j1 = j0 | (1 << p);
            float ar = sre[j0], ai = sim[j0];
            float br = sre[j1], bi = sim[j1];
            float g0r = g_re[gi][0], g0i = g_im[gi][0];
            float g1r = g_re[gi][1], g1i = g_im[gi][1];
            float g2r = g_re[gi][2], g2i = g_im[gi][2];
            float g3r = g_re[gi][3], g3i = g_im[gi][3];
            // v'[0] = g00*a + g01*b ; v'[1] = g10*a + g11*b   (complex)
            float n0r = g0r*ar - g0i*ai + g1r*br - g1i*bi;
            float n0i = g0r*ai + g0i*ar + g1r*bi + g1i*br;
            float n1r = g2r*ar - g2i*ai + g3r*br - g3i*bi;
            float n1i = g2r*ai + g2i*ar + g3r*bi + g3i*br;
            sre[j0] = n0r; sim[j0] = n0i;
            sre[j1] = n1r; sim[j1] = n1i;
            __syncthreads();
        }
        // ---- circular CNOTs: control q -> target (q+1)%NQ ----
        for (int q = 0; q < NQ; ++q) {
            const int cb = (NQ - 1) - q;
            const int tb = (NQ - 1) - ((q + 1) % NQ);
            if (t < 256) {
                const int hi = cb > tb ? cb : tb;
                const int lo = cb > tb ? tb : cb;
                int x = t;
                x = ((x >> lo) << (lo + 1)) | (x & ((1 << lo) - 1));
                x = ((x >> hi) << (hi + 1)) | (x & ((1 << hi) - 1));
                const int j0 = x | (1 << cb);       // control=1, target=0
                const int j1 = j0 | (1 << tb);      // control=1, target=1
                float ar = sre[j0], ai = sim[j0];
                float br = sre[j1], bi = sim[j1];
                sre[j0] = br; sim[j0] = bi;
                sre[j1] = ar; sim[j1] = ai;
            }
            __syncthreads();
        }
    }
    // U[row, j] = state[j]
    for (int j = t; j < NN; j += 512) {
        Ure[row * NN + j] = sre[j];
        Uim[row * NN + j] = sim[j];
    }
}

// ---------------- Kernel 2: fp32 WMMA GEMM, complex output ------------------
#define BM 64
#define BN 64
#define BK 32
#define LDK (BK + 2)   // padded LDS row stride: conflict-free, 8B-aligned v2f

__global__ __launch_bounds__(256)
void qgemm_wmma(const float* __restrict__ X,
                const float* __restrict__ Ure,
                const float* __restrict__ Uim,
                float* __restrict__ out) {
    __shared__ float As[BM * LDK];
    __shared__ float Br[BN * LDK];
    __shared__ float Bi[BN * LDK];

    const int tid  = threadIdx.x;
    const int lane = tid & 31;
    const int wave = tid >> 5;
    const int n0 = blockIdx.x * BN;
    const int m0 = blockIdx.y * BM;

    const int mw   = (wave >> 1) * 16;   // 4 M-groups of waves
    const int nw   = (wave & 1) * 32;    // 2 N-groups, 32 wide each
    const int half = lane >> 4;          // K sub-group per f32 WMMA layout
    const int lm   = lane & 15;          // M (A frag) / N (B frag) index

    v8f aR0 = {}; v8f aR1 = {}; v8f aI0 = {}; v8f aI1 = {};

    for (int kt = 0; kt < NN; kt += BK) {
        // Stage 64x32 tiles of X, Re(U), Im(U): 512 float4 per array, 2/thread
        #pragma unroll
        for (int it = 0; it < 2; ++it) {
            const int f  = tid + it * 256;
            const int r  = f >> 3;
            const int c4 = (f & 7) << 2;
            const float4 xa = *(const float4*)(X   + (size_t)(m0 + r) * NN + kt + c4);
            const float4 ur = *(const float4*)(Ure + (size_t)(n0 + r) * NN + kt + c4);
            const float4 ui = *(const float4*)(Uim + (size_t)(n0 + r) * NN + kt + c4);
            float* pa = As + r * LDK + c4;
            pa[0] = xa.x; pa[1] = xa.y; pa[2] = xa.z; pa[3] = xa.w;
            float* pr = Br + r * LDK + c4;
            pr[0] = ur.x; pr[1] = ur.y; pr[2] = ur.z; pr[3] = ur.w;
            float* pi = Bi + r * LDK + c4;
            pi[0] = ui.x; pi[1] = ui.y; pi[2] = ui.z; pi[3] = ui.w;
            if (kt + BK < NN) {   // prefetch next K slab -> global_prefetch_b8
                __builtin_prefetch(X   + (size_t)(m0 + r) * NN + kt + BK + c4, 0, 3);
                __builtin_prefetch(Ure + (size_t)(n0 + r) * NN + kt + BK + c4, 0, 3);
                __builtin_prefetch(Uim + (size_t)(n0 + r) * NN + kt + BK + c4, 0, 3);
            }
        }
        __syncthreads();

        #pragma unroll
        for (int kk = 0; kk < BK; kk += 4) {
            const int ka = kk + half * 2;   // VGPR0:K={0,2}, VGPR1:K={1,3}
            v2f a   = *(const v2f*)(As + (mw + lm)      * LDK + ka);
            v2f b0r = *(const v2f*)(Br + (nw + lm)      * LDK + ka);
            v2f b1r = *(const v2f*)(Br + (nw + 16 + lm) * LDK + ka);
            v2f b0i = *(const v2f*)(Bi + (nw + lm)      * LDK + ka);
            v2f b1i = *(const v2f*)(Bi + (nw + 16 + lm) * LDK + ka);
            aR0 = __builtin_amdgcn_wmma_f32_16x16x4_f32(false, a, false, b0r, (short)0, aR0, false, false);
            aR1 = __builtin_amdgcn_wmma_f32_16x16x4_f32(false, a, false, b1r, (short)0, aR1, false, false);
            aI0 = __builtin_amdgcn_wmma_f32_16x16x4_f32(false, a, false, b0i, (short)0, aI0, false, false);
            aI1 = __builtin_amdgcn_wmma_f32_16x16x4_f32(false, a, false, b1i, (short)0, aI1, false, false);
        }
        __syncthreads();
    }

    // Epilogue: C/D layout -> lane: N=lm, M=r+8*half. Interleave (re,im).
    #pragma unroll
    for (int r = 0; r < 8; ++r) {
        const int gm  = m0 + mw + r + 8 * half;
        const int gi0 = n0 + nw + lm;
        const int gi1 = gi0 + 16;
        float2 v0; v0.x = aR0[r]; v0.y = aI0[r];
        float2 v1; v1.x = aR1[r]; v1.y = aI1[r];
        *(float2*)(out + ((size_t)gm * NN + gi0) * 2) = v0;
        *(float2*)(out + ((size_t)gm * NN + gi1) * 2) = v1;
    }
}

// ---------------------------------------------------------------------------
extern "C" void kernel_launch(void* const* d_in, const int* in_sizes, int n_in,
                              void* d_out, int out_size, void* d_ws, size_t ws_size,
                              hipStream_t stream) {
    const float* x = (const float*)d_in[0];   // [2048, 1024] fp32
    const float* w = (const float*)d_in[1];   // [5, 10, 3]   fp32
    float* Ure = (float*)d_ws;                // 4 MB
    float* Uim = Ure + (size_t)NN * NN;       // 4 MB
    float* out = (float*)d_out;               // complex64 as float pairs

    build_u_rows<<<dim3(NN), dim3(512), 0, stream>>>(w, Ure, Uim);
    qgemm_wmma<<<dim3(BN == 64 ? NN / BN : 0, MT / BM), dim3(256), 0, stream>>>(x, Ure, Uim, out);
}